// SparseConv3d_23063974379959
// MI455X (gfx1250) — compile-verified
//
#include <hip/hip_runtime.h>

typedef __attribute__((ext_vector_type(2))) float v2f;
typedef __attribute__((ext_vector_type(8))) float v8f;

#define C_IN  32
#define C_OUT 32
#define K_OFF 27
// B-operand blocks: 27 offsets * 8 Cin-chunks * 2 Cout-tiles; each block = 32 lanes * float2
#define WBLK       (K_OFF * 16)
#define WLDS_BYTES (WBLK * 32 * 8)   // 110592 bytes of LDS

__global__ void spconv_zero_ws(float* ws) {
    if (threadIdx.x < 32) ws[threadIdx.x] = 0.0f;
}

__global__ __launch_bounds__(256)
void spconv_wmma_f32(const float* __restrict__ feats,
                     const float* __restrict__ weight,
                     const float* __restrict__ bias,
                     const long long* __restrict__ nbr,
                     float* __restrict__ out,
                     const float* __restrict__ zrow,
                     int n, int n_tiles)
{
    extern __shared__ float wlds[];

    const int tid  = threadIdx.x;
    const int lane = tid & 31;
    const int wv   = tid >> 5;
    const int m    = lane & 15;   // row (A) / col (B,D) within 16
    const int hi   = lane >> 4;   // half-wave select

    // ---- Stage weights into LDS, pre-permuted into WMMA B-operand layout ----
    // Block (k, c, t): lane L holds {W[k][4c+2*(L>>4)][16t+(L&15)],
    //                               W[k][4c+2*(L>>4)+1][16t+(L&15)]}
    for (int e = tid; e < WBLK * 32; e += 256) {
        const int blk = e >> 5;
        const int l   = e & 31;
        const int k   = blk >> 4;
        const int rem = blk & 15;
        const int c   = rem >> 1;
        const int t   = rem & 1;
        const int row0 = c * 4 + 2 * (l >> 4);
        const int col  = t * 16 + (l & 15);
        const float* wk = weight + k * (C_IN * C_OUT);
        v2f v;
        v.x = wk[row0 * C_OUT + col];
        v.y = wk[(row0 + 1) * C_OUT + col];
        *(v2f*)&wlds[e * 2] = v;
    }
    __syncthreads();

    const int tile = blockIdx.x * 8 + wv;
    if (tile >= n_tiles) return;      // wave-uniform: EXEC stays all-ones for WMMA

    const int vox0   = tile * 16;
    const int vrow   = vox0 + m;
    const int vclamp = vrow < n ? vrow : n - 1;
    const long long* idxp = nbr + (long long)vclamp * K_OFF;

    // D layout: VGPR r, lanes 0-15 -> M=r, N=lane; lanes 16-31 -> M=8+r, N=lane-16
    const float b0 = bias[m];
    const float b1 = bias[16 + m];
    v8f acc0, acc1;
    #pragma unroll
    for (int r = 0; r < 8; ++r) { acc0[r] = b0; acc1[r] = b1; }

    for (int k = 0; k < K_OFF; ++k) {
        const long long nb = idxp[k];
        // sentinel (nb == n) -> zero row in workspace; no EXEC divergence
        const float* src = (nb < (long long)n) ? (feats + nb * C_IN) : zrow;

        // A 16x4 f32 layout: lanes 0-15 K=0,1 ; lanes 16-31 K=2,3 of each chunk
        v2f a[8];
        #pragma unroll
        for (int c = 0; c < 8; ++c)
            a[c] = *(const v2f*)(src + c * 4 + 2 * hi);

        const float* wl = wlds + k * (16 * 64);
        #pragma unroll
        for (int c = 0; c < 8; ++c) {
            const v2f bt0 = *(const v2f*)(wl + (c * 2 + 0) * 64 + lane * 2);
            const v2f bt1 = *(const v2f*)(wl + (c * 2 + 1) * 64 + lane * 2);
            acc0 = __builtin_amdgcn_wmma_f32_16x16x4_f32(
                false, a[c], false, bt0, (short)0, acc0, false, false);
            acc1 = __builtin_amdgcn_wmma_f32_16x16x4_f32(
                false, a[c], false, bt1, (short)0, acc1, false, false);
        }
    }

    #pragma unroll
    for (int r = 0; r < 8; ++r) {
        const int row = vox0 + r + 8 * hi;
        if (row < n) {
            out[row * C_OUT + m]      = acc0[r];
            out[row * C_OUT + 16 + m] = acc1[r];
        }
    }
}

extern "C" void kernel_launch(void* const* d_in, const int* in_sizes, int n_in,
                              void* d_out, int out_size, void* d_ws, size_t ws_size,
                              hipStream_t stream) {
    (void)n_in; (void)out_size; (void)ws_size;
    const float*     feats  = (const float*)d_in[0];
    const float*     weight = (const float*)d_in[1];
    const float*     bias   = (const float*)d_in[2];
    const long long* nbr    = (const long long*)d_in[3];
    float* out  = (float*)d_out;
    float* zrow = (float*)d_ws;

    const int n       = in_sizes[0] / C_IN;
    const int n_tiles = (n + 15) / 16;

    spconv_zero_ws<<<1, 64, 0, stream>>>(zrow);

    const int blocks = (n_tiles + 7) / 8;   // 8 waves (tiles) per 256-thread block
    spconv_wmma_f32<<<blocks, 256, WLDS_BYTES, stream>>>(
        feats, weight, bias, nbr, out, zrow, n, n_tiles);
}